// HeteroGCNConv_59854664237638
// MI455X (gfx1250) — compile-verified
//
#include <hip/hip_runtime.h>

typedef float v2f __attribute__((ext_vector_type(2)));
typedef float v8f __attribute__((ext_vector_type(8)));

constexpr int DIMS = 128;  // D == OUT == 128

// ---------------------------------------------------------------- utilities
__global__ void zero_kernel(float* __restrict__ p, long long n) {
  long long i = (long long)blockIdx.x * blockDim.x + threadIdx.x;
  long long stride = (long long)gridDim.x * blockDim.x;
  for (; i < n; i += stride) p[i] = 0.0f;
}

__global__ void relu_kernel(float* __restrict__ p, long long n) {
  long long i = (long long)blockIdx.x * blockDim.x + threadIdx.x;
  long long stride = (long long)gridDim.x * blockDim.x;
  for (; i < n; i += stride) p[i] = fmaxf(p[i], 0.0f);
}

// Degree histogram via f32 atomic increments (avg degree ~10-20 -> low contention).
__global__ void degree_kernel(const int* __restrict__ idx,
                              float* __restrict__ deg, int n) {
  int i = blockIdx.x * blockDim.x + threadIdx.x;
  if (i < n) atomicAdd(&deg[idx[i]], 1.0f);
}

// ---------------------------------------------------------------- WMMA GEMM
// H[nrows x 128] = X[nrows x 128] @ W[128 x 128], fp32 via V_WMMA_F32_16X16X4_F32.
// 256 threads = 8 waves per block; W staged in LDS (64KB); each wave computes one
// 16-row tile across all 8 N-tiles with a K-loop of 32 steps (K=4 per WMMA).
__global__ __launch_bounds__(256) void gemm128_wmma_f32(
    const float* __restrict__ X, const float* __restrict__ W,
    float* __restrict__ H, int ntiles) {
  __shared__ float Wl[DIMS * DIMS];
  {
    const float4* Wv = (const float4*)W;
    float4* Wlv = (float4*)Wl;
    for (int i = threadIdx.x; i < (DIMS * DIMS) / 4; i += 256) Wlv[i] = Wv[i];
  }
  __syncthreads();

  const int wave = threadIdx.x >> 5;
  const int lane = threadIdx.x & 31;
  const int tile = blockIdx.x * 8 + wave;
  if (tile >= ntiles) return;  // wave-uniform; EXEC stays all-ones for WMMA

  const int m = lane & 15;           // row within tile (A) / col within tile (B,D)
  const int khalf = (lane >> 4) << 1;  // lanes 16-31 hold K+2,K+3

  const float* __restrict__ Xrow = X + ((long long)tile * 16 + m) * DIMS;

  v8f acc[8] = {};

  for (int k0 = 0; k0 < DIMS; k0 += 4) {
    // A fragment (16x4 f32): VGPR0 = K=k0|k0+2, VGPR1 = K=k0+1|k0+3
    v2f a;
    a.x = Xrow[k0 + khalf];
    a.y = Xrow[k0 + khalf + 1];
    const float* Wk0 = &Wl[(k0 + khalf) * DIMS];
    const float* Wk1 = &Wl[(k0 + khalf + 1) * DIMS];
#pragma unroll
    for (int t = 0; t < 8; ++t) {
      // B fragment (4x16 f32), mirrored layout
      const int n = t * 16 + m;
      v2f b;
      b.x = Wk0[n];
      b.y = Wk1[n];
      acc[t] = __builtin_amdgcn_wmma_f32_16x16x4_f32(
          false, a, false, b, (short)0, acc[t], false, false);
    }
  }

  // D layout: VGPR j -> lanes 0-15: M=j, N=lane; lanes 16-31: M=j+8, N=lane-16
  const int rowoff = (lane >> 4) << 3;
  const long long base = ((long long)tile * 16 + rowoff) * DIMS;
#pragma unroll
  for (int t = 0; t < 8; ++t) {
#pragma unroll
    for (int j = 0; j < 8; ++j) {
      H[base + (long long)j * DIMS + t * 16 + m] = acc[t][j];
    }
  }
}

// ---------------------------------------------------------------- scatter
// One wave32 per edge: each lane owns a float4 slice of the 128-wide message row
// (32 lanes x 16B = 512B coalesced), scaled by scale/sqrt(deg_s*deg_t), atomically
// accumulated into out[dst]. The x0.5 for node-type 'a' is folded into `scale`.
__global__ __launch_bounds__(256) void scatter_kernel(
    const float* __restrict__ H, const float* __restrict__ deg_s,
    const float* __restrict__ deg_t, const int* __restrict__ src,
    const int* __restrict__ dst, float* __restrict__ out, int nedges,
    float scale) {
  const int gw = (int)((blockIdx.x * 256u + threadIdx.x) >> 5);
  const int lane = threadIdx.x & 31;
  if (gw >= nedges) return;
  const int s = src[gw];
  const int d = dst[gw];
  const float inv = scale / sqrtf(deg_s[s] * deg_t[d]);
  const float4 v = ((const float4*)(H + (long long)s * DIMS))[lane];
  float* o = out + (long long)d * DIMS + lane * 4;
  atomicAdd(o + 0, v.x * inv);
  atomicAdd(o + 1, v.y * inv);
  atomicAdd(o + 2, v.z * inv);
  atomicAdd(o + 3, v.w * inv);
}

// ---------------------------------------------------------------- launch
extern "C" void kernel_launch(void* const* d_in, const int* in_sizes, int n_in,
                              void* d_out, int out_size, void* d_ws,
                              size_t ws_size, hipStream_t stream) {
  const float* x_a = (const float*)d_in[0];
  const float* x_b = (const float*)d_in[1];
  const float* W_ab = (const float*)d_in[2];
  const float* W_ba = (const float*)d_in[3];
  const float* W_aa = (const float*)d_in[4];
  const int* src_ab = (const int*)d_in[5];
  const int* dst_ab = (const int*)d_in[6];
  const int* src_ba = (const int*)d_in[7];
  const int* dst_ba = (const int*)d_in[8];
  const int* src_aa = (const int*)d_in[9];
  const int* dst_aa = (const int*)d_in[10];

  const int NA = in_sizes[0] / DIMS;  // 100000
  const int NB = in_sizes[1] / DIMS;  // 50000
  const int E = in_sizes[5];          // 1000000

  float* out = (float*)d_out;
  float* out_a = out;                        // [NA, 128]
  float* out_b = out + (long long)NA * DIMS; // [NB, 128]

  // workspace layout: H_ab | H_ba | H_aa | 6 degree arrays
  float* ws = (float*)d_ws;
  float* H_ab = ws;
  float* H_ba = H_ab + (long long)NA * DIMS;
  float* H_aa = H_ba + (long long)NB * DIMS;
  float* dsrc_ab = H_aa + (long long)NA * DIMS;  // [NA]
  float* ddst_ab = dsrc_ab + NA;                 // [NB]
  float* dsrc_ba = ddst_ab + NB;                 // [NB]
  float* ddst_ba = dsrc_ba + NB;                 // [NA]
  float* dsrc_aa = ddst_ba + NA;                 // [NA]
  float* ddst_aa = dsrc_aa + NA;                 // [NA]

  const long long out_total = (long long)out_size;
  const long long deg_total = 4LL * NA + 2LL * NB;

  // 1) zero accumulators (harness poisons d_out/d_ws)
  zero_kernel<<<2048, 256, 0, stream>>>(out, out_total);
  zero_kernel<<<512, 256, 0, stream>>>(dsrc_ab, deg_total);

  // 2) degrees
  const int degBlocks = (E + 255) / 256;
  degree_kernel<<<degBlocks, 256, 0, stream>>>(src_ab, dsrc_ab, E);
  degree_kernel<<<degBlocks, 256, 0, stream>>>(dst_ab, ddst_ab, E);
  degree_kernel<<<degBlocks, 256, 0, stream>>>(src_ba, dsrc_ba, E);
  degree_kernel<<<degBlocks, 256, 0, stream>>>(dst_ba, ddst_ba, E);
  degree_kernel<<<degBlocks, 256, 0, stream>>>(src_aa, dsrc_aa, E);
  degree_kernel<<<degBlocks, 256, 0, stream>>>(dst_aa, ddst_aa, E);

  // 3) node projections (fp32 WMMA); NA, NB are multiples of 16
  const int tilesA = NA / 16, tilesB = NB / 16;
  gemm128_wmma_f32<<<(tilesA + 7) / 8, 256, 0, stream>>>(x_a, W_ab, H_ab, tilesA);
  gemm128_wmma_f32<<<(tilesB + 7) / 8, 256, 0, stream>>>(x_b, W_ba, H_ba, tilesB);
  gemm128_wmma_f32<<<(tilesA + 7) / 8, 256, 0, stream>>>(x_a, W_aa, H_aa, tilesA);

  // 4) gather/normalize/scatter-add (one wave per edge; 8 waves per block)
  const int scBlocks = (E + 7) / 8;
  scatter_kernel<<<scBlocks, 256, 0, stream>>>(H_ab, dsrc_ab, ddst_ab, src_ab,
                                               dst_ab, out_b, E, 1.0f);
  scatter_kernel<<<scBlocks, 256, 0, stream>>>(H_ba, dsrc_ba, ddst_ba, src_ba,
                                               dst_ba, out_a, E, 0.5f);
  scatter_kernel<<<scBlocks, 256, 0, stream>>>(H_aa, dsrc_aa, ddst_aa, src_aa,
                                               dst_aa, out_a, E, 0.5f);

  // 5) final ReLU over both outputs
  relu_kernel<<<2048, 256, 0, stream>>>(out, out_total);
}